// GCN_49005576848207
// MI455X (gfx1250) — compile-verified
//
#include <hip/hip_runtime.h>
#include <hip/hip_bf16.h>
#include <math.h>

#define N_NODES 50000
#define N_EDGES 800000

typedef __attribute__((ext_vector_type(2))) float v2f;
typedef __attribute__((ext_vector_type(8))) float v8f;

// ---------------- utility fills / degree ----------------
__global__ void fill_kernel(float* __restrict__ p, float v, int n) {
  int i = blockIdx.x * blockDim.x + threadIdx.x;
  if (i < n) p[i] = v;
}

__global__ void deg_kernel(const int* __restrict__ dst, float* __restrict__ deg, int nE) {
  int e = blockIdx.x * blockDim.x + threadIdx.x;
  if (e < nE) unsafeAtomicAdd(&deg[dst[e]], 1.0f);
}

__global__ void dinv_kernel(float* __restrict__ d, int n) {
  int i = blockIdx.x * blockDim.x + threadIdx.x;
  if (i < n) {
    float v = d[i];
    d[i] = v > 0.0f ? rsqrtf(v) : 0.0f;
  }
}

// ---------------- fp32 WMMA GEMM: H[M x NCOLS] = X[M x 128] * W[128 x NCOLS] ----
// One wave computes one 16x16 tile via V_WMMA_F32_16X16X4_F32 (K-step 4).
// Block = (NCOLS/16) waves, covers 16 rows x NCOLS cols. A tile staged in LDS.
template <int NCOLS>
__global__ __launch_bounds__(NCOLS * 2) void gemm_wmma_kernel(
    const float* __restrict__ X, const float* __restrict__ W,
    float* __restrict__ H, int M) {
  constexpr int K = 128;
  constexpr int NW = NCOLS / 16;     // waves per block
  constexpr int NT = NW * 32;        // threads per block
  __shared__ float As[16][K + 4];    // stride 132 floats -> conflict-free column reads

  const int tid  = threadIdx.x;
  const int wave = tid >> 5;
  const int lane = tid & 31;
  const int row0 = blockIdx.x * 16;

  // cooperative, coalesced load of the 16x128 A tile
  for (int idx = tid; idx < 16 * K; idx += NT) {
    int r = idx >> 7, c = idx & 127;
    int gr = row0 + r;
    As[r][c] = (gr < M) ? X[(size_t)gr * K + c] : 0.0f;
  }
  __syncthreads();

  const int col0 = wave * 16;
  const int mn   = lane & 15;        // A row within tile == B/D column within tile
  const int kb   = (lane >> 4) * 2;  // lanes 0-15: K={0,1}; lanes 16-31: K={2,3}

  v8f c = {};
  #pragma unroll
  for (int k = 0; k < K; k += 4) {
    v2f a;
    a.x = As[mn][k + kb];
    a.y = As[mn][k + kb + 1];
    v2f b;
    const float* wp = W + (size_t)(k + kb) * NCOLS + col0 + mn;  // L2-resident
    b.x = wp[0];
    b.y = wp[NCOLS];
    // (neg_a, A, neg_b, B, c_mod, C, reuse_a, reuse_b)
    c = __builtin_amdgcn_wmma_f32_16x16x4_f32(false, a, false, b, (short)0, c,
                                              false, false);
  }

  // C/D layout: VGPR v -> row v (lanes 0-15) / row v+8 (lanes 16-31), col = lane&15
  const int rhalf = (lane >> 4) * 8;
  float* out = H + (size_t)row0 * NCOLS + col0 + mn;
  #pragma unroll
  for (int v = 0; v < 8; ++v) {
    int gr = row0 + rhalf + v;
    if (gr < M) out[(size_t)(rhalf + v) * NCOLS] = c[v];
  }
}

// ---------------- edge scatter: agg[dst] += h[src] * dinv[src]*dinv[dst] -------
// One wave per edge; each lane handles F/32 consecutive floats.
template <int F>
__global__ void scatter_kernel(const float* __restrict__ h,
                               const int* __restrict__ src,
                               const int* __restrict__ dst,
                               const float* __restrict__ dinv,
                               float* __restrict__ agg, int nE) {
  constexpr int V = F / 32;
  int gid  = blockIdx.x * blockDim.x + threadIdx.x;
  int e    = gid >> 5;
  int lane = gid & 31;
  if (e >= nE) return;
  int s = src[e], d = dst[e];
  float nrm = dinv[s] * dinv[d];
  const float* hp = h   + (size_t)s * F + lane * V;
  float*       op = agg + (size_t)d * F + lane * V;
  float vals[V];
  #pragma unroll
  for (int j = 0; j < V; ++j) vals[j] = hp[j];    // merges to b128/b64 load
  #pragma unroll
  for (int j = 0; j < V; ++j) unsafeAtomicAdd(&op[j], vals[j] * nrm);
}

// ---------------- layer-1 epilogue: self-loop + bias + ReLU (in place) --------
__global__ void epi1_kernel(float* __restrict__ a, const float* __restrict__ h,
                            const float* __restrict__ dinv,
                            const float* __restrict__ bias, int n) {
  int i = blockIdx.x * blockDim.x + threadIdx.x;
  if (i >= n) return;
  int node = i >> 7, c = i & 127;
  float di = dinv[node];
  float v = a[i] + h[i] * di * di + bias[c];
  a[i] = fmaxf(v, 0.0f);
}

// ---------------- layer-2 epilogue: self-loop + bias + log_softmax (64 cols) --
// One wave per node, 2 columns per lane. In-place on d_out.
__global__ void epi2_logsoftmax_kernel(float* __restrict__ out,
                                       const float* __restrict__ h2,
                                       const float* __restrict__ dinv,
                                       const float* __restrict__ bias,
                                       int nNodes) {
  int gid  = blockIdx.x * blockDim.x + threadIdx.x;
  int node = gid >> 5;
  int lane = gid & 31;
  if (node >= nNodes) return;
  float di = dinv[node];
  float d2 = di * di;
  float*       row  = out + (size_t)node * 64;
  const float* hrow = h2  + (size_t)node * 64;
  float v0 = row[lane]      + hrow[lane]      * d2 + bias[lane];
  float v1 = row[lane + 32] + hrow[lane + 32] * d2 + bias[lane + 32];
  float mx = fmaxf(v0, v1);
  #pragma unroll
  for (int off = 16; off > 0; off >>= 1) mx = fmaxf(mx, __shfl_xor(mx, off, 32));
  float s = expf(v0 - mx) + expf(v1 - mx);
  #pragma unroll
  for (int off = 16; off > 0; off >>= 1) s += __shfl_xor(s, off, 32);
  float lse = mx + logf(s);
  row[lane]      = v0 - lse;
  row[lane + 32] = v1 - lse;
}

// ---------------------------------------------------------------------------
extern "C" void kernel_launch(void* const* d_in, const int* in_sizes, int n_in,
                              void* d_out, int out_size, void* d_ws, size_t ws_size,
                              hipStream_t stream) {
  const float* x  = (const float*)d_in[0];
  const int*   ei = (const int*)  d_in[1];   // edge_index [2, E]
  const float* W1 = (const float*)d_in[2];
  const float* b1 = (const float*)d_in[3];
  const float* W2 = (const float*)d_in[4];
  const float* b2 = (const float*)d_in[5];
  const int* src = ei;
  const int* dst = ei + N_EDGES;

  float* dinv = (float*)d_ws;                        // N_NODES
  float* h1   = dinv + N_NODES;                      // N*128
  float* a1   = h1   + (size_t)N_NODES * 128;        // N*128 (agg1, then ReLU'd in place)
  float* h2   = a1   + (size_t)N_NODES * 128;        // N*64
  float* agg2 = (float*)d_out;                       // N*64 (log_softmax in place)

  // degrees (init 1.0 for the self-loop) -> dinv
  fill_kernel<<<(N_NODES + 255) / 256, 256, 0, stream>>>(dinv, 1.0f, N_NODES);
  deg_kernel<<<(N_EDGES + 255) / 256, 256, 0, stream>>>(dst, dinv, N_EDGES);
  dinv_kernel<<<(N_NODES + 255) / 256, 256, 0, stream>>>(dinv, N_NODES);

  // ---- layer 1 ----
  gemm_wmma_kernel<128><<<(N_NODES + 15) / 16, 256, 0, stream>>>(x, W1, h1, N_NODES);
  fill_kernel<<<(N_NODES * 128 + 255) / 256, 256, 0, stream>>>(a1, 0.0f, N_NODES * 128);
  scatter_kernel<128><<<(N_EDGES * 32 + 255) / 256, 256, 0, stream>>>(h1, src, dst, dinv, a1, N_EDGES);
  epi1_kernel<<<(N_NODES * 128 + 255) / 256, 256, 0, stream>>>(a1, h1, dinv, b1, N_NODES * 128);

  // ---- layer 2 ----
  gemm_wmma_kernel<64><<<(N_NODES + 15) / 16, 128, 0, stream>>>(a1, W2, h2, N_NODES);
  fill_kernel<<<(N_NODES * 64 + 255) / 256, 256, 0, stream>>>(agg2, 0.0f, N_NODES * 64);
  scatter_kernel<64><<<(N_EDGES * 32 + 255) / 256, 256, 0, stream>>>(h2, src, dst, dinv, agg2, N_EDGES);
  epi2_logsoftmax_kernel<<<(N_NODES * 32 + 255) / 256, 256, 0, stream>>>(agg2, h2, dinv, b2, N_NODES);
}